// GAENodeClassificationEncoder_28767690948708
// MI455X (gfx1250) — compile-verified
//
#include <hip/hip_runtime.h>

// ---------------------------------------------------------------------------
// 2-layer GCN (GAE encoder) for MI455X / gfx1250, wave32.
//   h0 = emb[x]                                  (gather)
//   h1 = relu( Agg( h0 @ W1 ) + b1 )             (WMMA f32 GEMM + edge scatter)
//   out = Agg( h1 @ W2 ) + b2                    (WMMA f32 GEMM + edge scatter)
// Agg(m)[d] = sum_{e: dst(e)=d} inv[src]*inv[dst]*m[src] + inv[d]^2 * m[d]
//
// GEMM: V_WMMA_F32_16X16X4_F32, 8 waves/block, each wave = 16 M-rows x 128 N
// (8 accumulator tiles), B staged through LDS in 32-deep K slices via
// GLOBAL_LOAD_ASYNC_TO_LDS_B128 (ASYNCcnt) when available, ds_store_b128
// fallback otherwise.
//
// Workspace (bytes): inv 200000 | deg 200000 | h1 51.2MB | agg1 51.2MB
//   => ~98.1 MB total. h2 reuses the h1 region; h0 lives in d_out.
// ---------------------------------------------------------------------------

#define N_NODES 50000
#define N_EDGES 800000
#define EMB     128
#define H1DIM   256   // 2*HID
#define H2DIM   128   // HID

typedef __attribute__((ext_vector_type(2))) float v2f;
typedef __attribute__((ext_vector_type(8))) float v8f;

#if __has_builtin(__builtin_amdgcn_global_load_async_to_lds_b128) && \
    __has_builtin(__builtin_amdgcn_s_wait_asynccnt)
#define USE_ASYNC_LDS 1
#else
#define USE_ASYNC_LDS 0
#endif

// 128-bit payload type matching the builtin's parameter (per compiler diagnostic):
//   '__attribute__((__vector_size__(4 * sizeof(int)))) int *'
typedef int v4i __attribute__((vector_size(4 * sizeof(int))));
typedef __attribute__((address_space(1))) v4i* v4i_gptr;   // global
typedef __attribute__((address_space(3))) v4i* v4i_lptr;   // LDS

// -------------------- degree / normalization --------------------

__global__ void zero_deg_kernel(unsigned* __restrict__ deg) {
    int i = blockIdx.x * blockDim.x + threadIdx.x;
    if (i < N_NODES) deg[i] = 0u;
}

__global__ void count_deg_kernel(const int* __restrict__ dst, unsigned* __restrict__ deg) {
    int e = blockIdx.x * blockDim.x + threadIdx.x;
    if (e < N_EDGES) atomicAdd(&deg[dst[e]], 1u);
}

__global__ void inv_sqrt_kernel(const unsigned* __restrict__ deg, float* __restrict__ inv) {
    int i = blockIdx.x * blockDim.x + threadIdx.x;
    if (i < N_NODES) {
        float d = (float)(deg[i] + 1u);   // +1 self loop; always > 0
        inv[i] = 1.0f / sqrtf(d);
    }
}

// -------------------- embedding gather (float4 lanes) --------------------

__global__ void gather_emb_kernel(const int* __restrict__ x,
                                  const float* __restrict__ emb,
                                  float* __restrict__ h0) {
    int idx = blockIdx.x * blockDim.x + threadIdx.x;     // over N_NODES * 32
    if (idx >= N_NODES * (EMB / 4)) return;
    int node = idx >> 5;
    int q    = idx & 31;
    int v    = x[node];
    reinterpret_cast<float4*>(h0)[node * 32 + q] =
        reinterpret_cast<const float4*>(emb)[v * 32 + q];
}

// -------------------- WMMA f32 GEMM with LDS-staged B --------------------
// C[M x N] = A[M x K] @ B[K x N].  M % 16 == 0, K % KTILE == 0.
// Grid: (ceil(M/16/8), N/NW).  Block: 256 threads = 8 waves.
// Wave w computes M-tile (blockIdx.x*8 + w) x NW columns (NT=NW/16 accum tiles).
// Fragment layouts per CDNA5 ISA 7.12.2 (32-bit A 16x4 / C-D 16x16 tables).

template <int NW, int KTILE>
__global__ __launch_bounds__(256) void wmma_gemm_lds_kernel(
        const float* __restrict__ A, const float* __restrict__ B,
        float* __restrict__ C, int M, int N, int K) {
    constexpr int NT = NW / 16;
    __shared__ float Bs[KTILE * NW];           // 32x128 f32 = 16 KB

    const int tid  = threadIdx.x;
    const int wave = tid >> 5;
    const int lane = tid & 31;
    const int g    = lane >> 4;                // lane group (0: lanes 0-15, 1: 16-31)
    const int r    = lane & 15;

    const int Mtiles = M >> 4;
    int  mtile  = blockIdx.x * 8 + wave;
    const bool active = (mtile < Mtiles);
    if (!active) mtile = Mtiles - 1;           // clamp: redundant compute, no store;
                                               // keeps barriers convergent, EXEC all-1s
    const int n0 = blockIdx.y * NW;

    const float* __restrict__ Arow = A + (size_t)(mtile * 16 + r) * K;

    v8f acc[NT];
    #pragma unroll
    for (int t = 0; t < NT; ++t) acc[t] = (v8f){0.f,0.f,0.f,0.f,0.f,0.f,0.f,0.f};

    for (int k0 = 0; k0 < K; k0 += KTILE) {
        __syncthreads();                       // protect Bs from previous iteration
        // ---- stage B[k0 .. k0+KTILE) x [n0 .. n0+NW) into LDS ----
        #pragma unroll
        for (int c = tid; c < (KTILE * NW) / 4; c += 256) {
            const int kk = c / (NW / 4);
            const int cc = c - kk * (NW / 4);
            const float* gsrc = B + (size_t)(k0 + kk) * N + n0 + cc * 4;
#if USE_ASYNC_LDS
            __builtin_amdgcn_global_load_async_to_lds_b128(
                (v4i_gptr)gsrc, (v4i_lptr)&Bs[kk * NW + cc * 4], 0, 0);
#else
            *reinterpret_cast<float4*>(&Bs[kk * NW + cc * 4]) =
                *reinterpret_cast<const float4*>(gsrc);
#endif
        }
#if USE_ASYNC_LDS
        __builtin_amdgcn_s_wait_asynccnt(0);   // my copies landed in LDS
#endif
        __syncthreads();                       // everyone's copies visible

        // ---- compute: KTILE/4 WMMA steps, 1 A-pair load feeds NT WMMAs ----
        #pragma unroll
        for (int kk = 0; kk < KTILE; kk += 4) {
            const int ka = kk + 2 * g;         // K pair for this lane group
            v2f a = *reinterpret_cast<const v2f*>(Arow + k0 + ka);  // global_load_b64
            #pragma unroll
            for (int t = 0; t < NT; ++t) {
                v2f b;
                b.x = Bs[(ka    ) * NW + t * 16 + r];   // ds_load_b32
                b.y = Bs[(ka + 1) * NW + t * 16 + r];   // ds_load_b32
                // 8 args: (neg_a, A, neg_b, B, c_mod, C, reuse_a, reuse_b)
                acc[t] = __builtin_amdgcn_wmma_f32_16x16x4_f32(
                             false, a, false, b, (short)0, acc[t], false, false);
            }
        }
    }

    if (active) {
        // C/D layout: VGPR v, lanes 0-15 -> row v, lanes 16-31 -> row v+8; col = lane&15
        #pragma unroll
        for (int t = 0; t < NT; ++t) {
            float* Cb = C + (size_t)(mtile * 16 + 8 * g) * N + n0 + t * 16 + r;
            #pragma unroll
            for (int v = 0; v < 8; ++v) Cb[(size_t)v * N] = acc[t][v];
        }
    }
}

// -------------------- self-loop + bias init: out = b + inv^2 * h --------------------

__global__ void self_loop_init_kernel(const float* __restrict__ h,
                                      const float* __restrict__ bias,
                                      const float* __restrict__ inv,
                                      float* __restrict__ out, int fs) {
    int idx = blockIdx.x * blockDim.x + threadIdx.x;    // over N_NODES << fs
    if (idx >= (N_NODES << fs)) return;
    int node = idx >> fs;
    int j    = idx & ((1 << fs) - 1);
    float s  = inv[node];
    out[idx] = bias[j] + s * s * h[idx];
}

// -------------------- edge scatter: out[dst] += inv[s]*inv[d]*h[src] --------------------
// Flat (edge, feature) parallelization; consecutive lanes share one edge so
// src/dst/inv loads are broadcast and h/out accesses are fully coalesced.
// h1 (51 MB) and out (51 MB) both fit in the 192 MB L2, so the atomic stream
// runs at L2 rather than HBM rate.

__global__ void scatter_edges_kernel(const int* __restrict__ src,
                                     const int* __restrict__ dst,
                                     const float* __restrict__ inv,
                                     const float* __restrict__ h,
                                     float* __restrict__ out, int fs) {
    long long idx = (long long)blockIdx.x * blockDim.x + threadIdx.x;
    long long total = (long long)N_EDGES << fs;
    if (idx >= total) return;
    int e = (int)(idx >> fs);
    int j = (int)(idx & ((1 << fs) - 1));
    int s = src[e];
    int d = dst[e];
    float w = inv[s] * inv[d];
    atomicAdd(&out[((long long)d << fs) + j], w * h[((long long)s << fs) + j]);
}

// -------------------- ReLU --------------------

__global__ void relu_kernel(float* __restrict__ a, int n) {
    int i = blockIdx.x * blockDim.x + threadIdx.x;
    if (i < n) a[i] = fmaxf(a[i], 0.0f);
}

// ---------------------------------------------------------------------------

extern "C" void kernel_launch(void* const* d_in, const int* in_sizes, int n_in,
                              void* d_out, int out_size, void* d_ws, size_t ws_size,
                              hipStream_t stream) {
    (void)in_sizes; (void)n_in; (void)out_size; (void)ws_size;

    // Inputs in setup_inputs() order: x, edge_index, emb, W1, b1, W2, b2
    const int*   x    = (const int*)d_in[0];              // [N_NODES]
    const int*   eidx = (const int*)d_in[1];              // [2, N_EDGES] flat
    const float* emb  = (const float*)d_in[2];            // [4096, 128]
    const float* W1   = (const float*)d_in[3];            // [128, 256]
    const float* b1   = (const float*)d_in[4];            // [256]
    const float* W2   = (const float*)d_in[5];            // [256, 128]
    const float* b2   = (const float*)d_in[6];            // [128]
    const int* src = eidx;
    const int* dst = eidx + N_EDGES;

    // Workspace carve-up (256B aligned)
    char* ws = (char*)d_ws;
    float*    inv  = (float*)(ws);                        // 50000 f32
    unsigned* deg  = (unsigned*)(ws + 200192);            // 50000 u32
    float*    h1   = (float*)(ws + 400384);               // 50000 x 256 f32
    float*    agg1 = (float*)(ws + 400384 + 51200000);    // 50000 x 256 f32
    float*    h2   = h1;                                  // reuse (h1 dead after ReLU)
    float*    h0   = (float*)d_out;                       // 50000 x 128 f32 (scratch, overwritten)
    float*    out  = (float*)d_out;

    const int B = 256;
    const int mblocks = (N_NODES / 16 + 7) / 8;           // 3125 M-tiles -> 391 blocks

    // 1) degrees with self-loops -> inv_sqrt
    zero_deg_kernel<<<(N_NODES + B - 1) / B, B, 0, stream>>>(deg);
    count_deg_kernel<<<(N_EDGES + B - 1) / B, B, 0, stream>>>(dst, deg);
    inv_sqrt_kernel<<<(N_NODES + B - 1) / B, B, 0, stream>>>(deg, inv);

    // 2) h0 = emb[x]
    gather_emb_kernel<<<(N_NODES * (EMB / 4) + B - 1) / B, B, 0, stream>>>(x, emb, h0);

    // 3) h1 = h0 @ W1   (M=50000, K=128, N=256): 2 N-slabs of 128
    wmma_gemm_lds_kernel<128, 32><<<dim3(mblocks, H1DIM / 128), B, 0, stream>>>(
        h0, W1, h1, N_NODES, H1DIM, EMB);

    // 4) agg1 = b1 + inv^2*h1 ; += edge messages ; ReLU
    self_loop_init_kernel<<<(N_NODES * H1DIM) / B, B, 0, stream>>>(h1, b1, inv, agg1, 8);
    scatter_edges_kernel<<<(int)(((long long)N_EDGES << 8) / B), B, 0, stream>>>(src, dst, inv, h1, agg1, 8);
    relu_kernel<<<(N_NODES * H1DIM) / B, B, 0, stream>>>(agg1, N_NODES * H1DIM);

    // 5) h2 = agg1 @ W2  (M=50000, K=256, N=128): 1 N-slab
    wmma_gemm_lds_kernel<128, 32><<<dim3(mblocks, H2DIM / 128), B, 0, stream>>>(
        agg1, W2, h2, N_NODES, H2DIM, H1DIM);

    // 6) out = b2 + inv^2*h2 ; += edge messages
    self_loop_init_kernel<<<(N_NODES * H2DIM) / B, B, 0, stream>>>(h2, b2, inv, out, 7);
    scatter_edges_kernel<<<(int)(((long long)N_EDGES << 7) / B), B, 0, stream>>>(src, dst, inv, h2, out, 7);
}